// KGAT_58067957842412
// MI455X (gfx1250) — compile-verified
//
#include <hip/hip_runtime.h>
#include <math.h>

#define N_NODES 100000
#define DIM 64
#define SE 16
#define BATCH 4096
#define EDGES_PER_NODE (SE + 1)   // SE graph edges + 1 self loop
#define RULESCORE 0.1f
#define NEG_SLOPE 0.2f

typedef float v2f __attribute__((ext_vector_type(2)));
typedef float v8f __attribute__((ext_vector_type(8)));

// -------------------------------------------------------------------------
// K1: H = X @ W  via V_WMMA_F32_16X16X4_F32.
// Block = 256 threads = 8 waves; each wave computes a 16x64 tile of H.
// W (64x64 = 16KB) is staged in LDS cooperatively.
// A layout (16x4 f32): lane m (m<16) holds {K=kb, K=kb+1}, lane m+16 holds
// {K=kb+2, K=kb+3} where kb = 4*kstep. B layout (4x16): VGPR0 = rows kb/kb+2
// striped over lanes, VGPR1 = rows kb+1/kb+3. D: VGPR v = row v (+8 for hi half).
// -------------------------------------------------------------------------
__global__ __launch_bounds__(256) void gemm_xw_kernel(const float* __restrict__ X,
                                                      const float* __restrict__ W,
                                                      float* __restrict__ H) {
  __shared__ float Wlds[DIM * DIM];
  for (int i = threadIdx.x; i < DIM * DIM; i += 256) Wlds[i] = W[i];
  __syncthreads();

  const int lane = threadIdx.x & 31;
  const int wave = threadIdx.x >> 5;
  const int lo = lane & 15;
  const int hi = lane >> 4;
  const int rowBase = blockIdx.x * 128 + wave * 16;
  if (rowBase >= N_NODES) return;   // wave-uniform exit; EXEC stays all-ones

  const float* Xrow = X + (size_t)(rowBase + lo) * DIM;
  __builtin_prefetch(Xrow, 0, 0);   // global_prefetch_b8

  v8f acc0 = {}, acc1 = {}, acc2 = {}, acc3 = {};
#pragma unroll
  for (int k = 0; k < 16; ++k) {
    const int kb = 4 * k + 2 * hi;
    v2f a = *(const v2f*)(Xrow + kb);           // 8B-aligned pair load
    v2f b0, b1, b2, b3;
    b0.x = Wlds[kb * DIM + lo];        b0.y = Wlds[(kb + 1) * DIM + lo];
    b1.x = Wlds[kb * DIM + 16 + lo];   b1.y = Wlds[(kb + 1) * DIM + 16 + lo];
    b2.x = Wlds[kb * DIM + 32 + lo];   b2.y = Wlds[(kb + 1) * DIM + 32 + lo];
    b3.x = Wlds[kb * DIM + 48 + lo];   b3.y = Wlds[(kb + 1) * DIM + 48 + lo];
    acc0 = __builtin_amdgcn_wmma_f32_16x16x4_f32(false, a, false, b0, (short)0, acc0, false, false);
    acc1 = __builtin_amdgcn_wmma_f32_16x16x4_f32(false, a, false, b1, (short)0, acc1, false, false);
    acc2 = __builtin_amdgcn_wmma_f32_16x16x4_f32(false, a, false, b2, (short)0, acc2, false, false);
    acc3 = __builtin_amdgcn_wmma_f32_16x16x4_f32(false, a, false, b3, (short)0, acc3, false, false);
  }

#pragma unroll
  for (int v = 0; v < 8; ++v) {
    float* hp = H + (size_t)(rowBase + v + 8 * hi) * DIM;
    hp[lo]      = acc0[v];
    hp[16 + lo] = acc1[v];
    hp[32 + lo] = acc2[v];
    hp[48 + lo] = acc3[v];
  }
}

// -------------------------------------------------------------------------
// K2: a_i[n] = h[n].att_i ; a_j[n] = h[n].att_j
// -------------------------------------------------------------------------
__global__ void node_alpha_kernel(const float* __restrict__ H,
                                  const float* __restrict__ att,
                                  float* __restrict__ ai, float* __restrict__ aj) {
  int n = blockIdx.x * blockDim.x + threadIdx.x;
  if (n >= N_NODES) return;
  const float4* h4 = (const float4*)(H + (size_t)n * DIM);
  const float4* ai4 = (const float4*)(att);
  const float4* aj4 = (const float4*)(att + DIM);
  float si = 0.f, sj = 0.f;
#pragma unroll
  for (int i = 0; i < DIM / 4; ++i) {
    float4 v = h4[i], wi = ai4[i], wj = aj4[i];
    si += v.x * wi.x + v.y * wi.y + v.z * wi.z + v.w * wi.w;
    sj += v.x * wj.x + v.y * wj.y + v.z * wj.z + v.w * wj.w;
  }
  ai[n] = si; aj[n] = sj;
}

// -------------------------------------------------------------------------
// K3: init scratch (g accumulator, segment max, denom) and reg-loss cell
// -------------------------------------------------------------------------
__global__ void init_kernel(float* __restrict__ G, float* __restrict__ amax,
                            float* __restrict__ denom, float* __restrict__ out) {
  long long i = (long long)blockIdx.x * blockDim.x + threadIdx.x;
  if (i < (long long)N_NODES * DIM) G[i] = 0.f;
  if (i < N_NODES) { amax[i] = -1e30f; denom[i] = 0.f; }
  if (i == 0) out[2 * BATCH] = 0.f;
}

__device__ __forceinline__ float leaky(float x) { return x > 0.f ? x : NEG_SLOPE * x; }

// -------------------------------------------------------------------------
// K4: segment max over dst  (global_atomic_max_num_f32)
// edge e: n = e/17 is src; k<16 -> dst = E[n*16+k] (invalid if ==n); k==16 self loop
// -------------------------------------------------------------------------
__global__ void edge_max_kernel(const int* __restrict__ E,
                                const float* __restrict__ ai,
                                const float* __restrict__ aj,
                                float* __restrict__ amax) {
  long long e = (long long)blockIdx.x * blockDim.x + threadIdx.x;
  if (e >= (long long)N_NODES * EDGES_PER_NODE) return;
  int n = (int)(e / EDGES_PER_NODE);
  int k = (int)(e - (long long)n * EDGES_PER_NODE);
  int m;
  if (k == SE) m = n;
  else { m = E[n * SE + k]; if (m == n) return; }
  atomicMax(&amax[m], leaky(ai[m] + aj[n]));
}

// -------------------------------------------------------------------------
// K5: e = exp(alpha - amax[dst]); denom[dst] += e; stash e in coef[edge]
// -------------------------------------------------------------------------
__global__ void edge_exp_kernel(const int* __restrict__ E,
                                const float* __restrict__ ai,
                                const float* __restrict__ aj,
                                const float* __restrict__ amax,
                                float* __restrict__ denom,
                                float* __restrict__ coef) {
  long long e = (long long)blockIdx.x * blockDim.x + threadIdx.x;
  if (e >= (long long)N_NODES * EDGES_PER_NODE) return;
  int n = (int)(e / EDGES_PER_NODE);
  int k = (int)(e - (long long)n * EDGES_PER_NODE);
  int m;
  if (k == SE) m = n;
  else { m = E[n * SE + k]; if (m == n) { coef[e] = 0.f; return; } }
  float v = expf(leaky(ai[m] + aj[n]) - amax[m]);
  coef[e] = v;
  atomicAdd(&denom[m], v);
}

// -------------------------------------------------------------------------
// K6: coef[edge] /= (denom[dst] + 1e-16)
// -------------------------------------------------------------------------
__global__ void edge_div_kernel(const int* __restrict__ E,
                                const float* __restrict__ denom,
                                float* __restrict__ coef) {
  long long e = (long long)blockIdx.x * blockDim.x + threadIdx.x;
  if (e >= (long long)N_NODES * EDGES_PER_NODE) return;
  int n = (int)(e / EDGES_PER_NODE);
  int k = (int)(e - (long long)n * EDGES_PER_NODE);
  int m = (k == SE) ? n : E[n * SE + k];       // invalid edges have coef==0
  coef[e] = coef[e] / (denom[m] + 1e-16f);
}

// -------------------------------------------------------------------------
// K7: g[dst] += coef * h[src]   (64 lanes per edge, f32 scatter-add; g fits L2)
// -------------------------------------------------------------------------
__global__ void edge_agg_kernel(const int* __restrict__ E,
                                const float* __restrict__ coef,
                                const float* __restrict__ H,
                                float* __restrict__ G) {
  long long t = (long long)blockIdx.x * blockDim.x + threadIdx.x;
  if (t >= (long long)N_NODES * EDGES_PER_NODE * DIM) return;
  long long e = t >> 6;
  int d = (int)(t & 63);
  int n = (int)(e / EDGES_PER_NODE);
  int k = (int)(e - (long long)n * EDGES_PER_NODE);
  float c = coef[e];
  if (c == 0.f) return;
  int m = (k == SE) ? n : E[n * SE + k];
  atomicAdd(&G[(size_t)m * DIM + d], H[(size_t)n * DIM + d] * c);
}

// -------------------------------------------------------------------------
// K8: g = (g + bias) / max(||g + bias||, 1e-12)
// -------------------------------------------------------------------------
__global__ void normalize_kernel(float* __restrict__ G, const float* __restrict__ bias) {
  int n = blockIdx.x * blockDim.x + threadIdx.x;
  if (n >= N_NODES) return;
  float* g = G + (size_t)n * DIM;
  float vals[DIM];
  float ss = 0.f;
#pragma unroll
  for (int d = 0; d < DIM; ++d) { float v = g[d] + bias[d]; vals[d] = v; ss += v * v; }
  float inv = 1.f / fmaxf(sqrtf(ss), 1e-12f);
#pragma unroll
  for (int d = 0; d < DIM; ++d) g[d] = vals[d] * inv;
}

// -------------------------------------------------------------------------
// K9: scoring.  u_e = concat(AE[u], G[u]) (128-d). One thread per batch row.
// -------------------------------------------------------------------------
__device__ __forceinline__ float dot64(const float* __restrict__ a,
                                       const float* __restrict__ b) {
  const float4* A = (const float4*)a;
  const float4* Bv = (const float4*)b;
  float s = 0.f;
#pragma unroll
  for (int i = 0; i < DIM / 4; ++i) {
    float4 x = A[i], y = Bv[i];
    s += x.x * y.x + x.y * y.y + x.z * y.z + x.w * y.w;
  }
  return s;
}

__global__ __launch_bounds__(256) void score_kernel(const float* __restrict__ AE,
                                                    const float* __restrict__ G,
                                                    const int* __restrict__ user,
                                                    const int* __restrict__ pos,
                                                    const int* __restrict__ neg,
                                                    const int* __restrict__ posr,
                                                    const int* __restrict__ negr,
                                                    float* __restrict__ out) {
  int b = blockIdx.x * 256 + threadIdx.x;
  float regp = 0.f;
  if (b < BATCH) {
    int u = user[b], p = pos[b], q = neg[b], pr = posr[b], nr = negr[b];
    const float* ua = AE + (size_t)u * DIM; const float* ug = G + (size_t)u * DIM;
    const float* pa = AE + (size_t)p * DIM; const float* pg = G + (size_t)p * DIM;
    const float* qa = AE + (size_t)q * DIM; const float* qg = G + (size_t)q * DIM;
    const float* ra = AE + (size_t)pr * DIM; const float* rg = G + (size_t)pr * DIM;
    const float* sa = AE + (size_t)nr * DIM; const float* sg = G + (size_t)nr * DIM;

    float s_up  = dot64(ua, pa) + dot64(ug, pg);
    float s_upr = dot64(ua, ra) + dot64(ug, rg);
    float s_un  = dot64(ua, qa) + dot64(ug, qg);
    float s_unr = dot64(ua, sa) + dot64(ug, sg);
    out[b]         = s_up + RULESCORE * s_upr;
    out[BATCH + b] = s_un + RULESCORE * s_unr;

    float n2u = dot64(ua, ua) + dot64(ug, ug);
    float n2p = dot64(pa, pa) + dot64(pg, pg);
    float n2q = dot64(qa, qa) + dot64(qg, qg);
    float n2r = dot64(ra, ra) + dot64(rg, rg);
    float n2s = dot64(sa, sa) + dot64(sg, sg);
    regp = 0.5f * (n2u + n2p + n2q) + 0.0005f * (n2r + n2s);
  }
  __shared__ float red[256];
  red[threadIdx.x] = regp;
  __syncthreads();
  for (int s = 128; s > 0; s >>= 1) {
    if (threadIdx.x < s) red[threadIdx.x] += red[threadIdx.x + s];
    __syncthreads();
  }
  if (threadIdx.x == 0) atomicAdd(&out[2 * BATCH], red[0]);
}

// -------------------------------------------------------------------------
extern "C" void kernel_launch(void* const* d_in, const int* in_sizes, int n_in,
                              void* d_out, int out_size, void* d_ws, size_t ws_size,
                              hipStream_t stream) {
  const float* AE   = (const float*)d_in[0];  // all_embed  N x 64
  const float* W    = (const float*)d_in[1];  // weight     64 x 64
  const float* att  = (const float*)d_in[2];  // att        128
  const float* bias = (const float*)d_in[3];  // bias       64
  const int*   E    = (const int*)d_in[4];    // edges      N x 16
  const int*   user = (const int*)d_in[5];
  const int*   pos  = (const int*)d_in[6];
  const int*   neg  = (const int*)d_in[7];
  const int*   posr = (const int*)d_in[8];
  const int*   negr = (const int*)d_in[9];
  float* out = (float*)d_out;

  float* H     = (float*)d_ws;               // N*64
  float* G     = H + (size_t)N_NODES * DIM;  // N*64
  float* ai    = G + (size_t)N_NODES * DIM;  // N
  float* aj    = ai + N_NODES;               // N
  float* amax  = aj + N_NODES;               // N
  float* denom = amax + N_NODES;             // N
  float* coef  = denom + N_NODES;            // N*17

  const long long nEdges = (long long)N_NODES * EDGES_PER_NODE;

  // init (covers N*64 range)
  {
    long long total = (long long)N_NODES * DIM;
    int blocks = (int)((total + 255) / 256);
    init_kernel<<<blocks, 256, 0, stream>>>(G, amax, denom, out);
  }
  // GEMM: 128 rows per block
  gemm_xw_kernel<<<(N_NODES + 127) / 128, 256, 0, stream>>>(AE, W, H);
  node_alpha_kernel<<<(N_NODES + 255) / 256, 256, 0, stream>>>(H, att, ai, aj);
  {
    int blocks = (int)((nEdges + 255) / 256);
    edge_max_kernel<<<blocks, 256, 0, stream>>>(E, ai, aj, amax);
    edge_exp_kernel<<<blocks, 256, 0, stream>>>(E, ai, aj, amax, denom, coef);
    edge_div_kernel<<<blocks, 256, 0, stream>>>(E, denom, coef);
  }
  {
    long long total = nEdges * DIM;
    int blocks = (int)((total + 255) / 256);
    edge_agg_kernel<<<blocks, 256, 0, stream>>>(E, coef, H, G);
  }
  normalize_kernel<<<(N_NODES + 255) / 256, 256, 0, stream>>>(G, bias);
  score_kernel<<<(BATCH + 255) / 256, 256, 0, stream>>>(AE, G, user, pos, neg, posr, negr, out);
}